// TransformerWrapper_53257594471013
// MI455X (gfx1250) — compile-verified
//
#include <hip/hip_runtime.h>

typedef __attribute__((ext_vector_type(2))) float v2f;
typedef __attribute__((ext_vector_type(8))) float v8f;
typedef __attribute__((ext_vector_type(4))) int   v4i;

#define HEADS 8
#define OUTC  32
#define HC    256
#define ATT_SCALE 0.17677669529663687f  /* 1/sqrt(32) */

#define WMMA_F32(a, b, c) \
    __builtin_amdgcn_wmma_f32_16x16x4_f32(false, (a), false, (b), (short)0, (c), false, false)

typedef __attribute__((address_space(1))) v4i* gvec_t;   // global int4*
typedef __attribute__((address_space(3))) v4i* lvec_t;   // LDS int4*

// 16-byte async copy global -> LDS (CDNA5 GLOBAL_LOAD_ASYNC_TO_LDS_B128)
__device__ __forceinline__ void async_copy16(const float* g, float* l)
{
#if __has_builtin(__builtin_amdgcn_global_load_async_to_lds_b128)
    __builtin_amdgcn_global_load_async_to_lds_b128((gvec_t)g, (lvec_t)l, 0, 0);
#else
    const unsigned loff = (unsigned)(uintptr_t)l;
    asm volatile("global_load_async_to_lds_b128 %0, %1, off"
                 :: "v"(loff), "v"(g) : "memory");
#endif
}

__device__ __forceinline__ void wait_async0()
{
#if __has_builtin(__builtin_amdgcn_s_wait_asynccnt)
    __builtin_amdgcn_s_wait_asynccnt(0);
#else
    asm volatile("s_wait_asynccnt 0x0" ::: "memory");
#endif
}

// ---------------------------------------------------------------------------
// Generic fp32 WMMA GEMM:  C[M x 256] = A[M x Kdim] @ W[256 x Kdim]^T (+ bias)
// Block = 4 waves; the shared 16 x Kdim A tile (contiguous in memory) is
// async-staged into LDS once (stride-padded, conflict-free), then each wave
// computes 16 rows x 64 cols with B register-pipelined one K-step ahead.
// ---------------------------------------------------------------------------
__global__ __launch_bounds__(128)
void gemm16_wmma(const float* __restrict__ A, const float* __restrict__ W,
                 const float* __restrict__ bias, float* __restrict__ C,
                 int Kdim, int ldc)
{
    __shared__ float lds_a[16 * (HC + 4)];
    const int lstr = Kdim + 4;            // padded row stride (floats)

    const int lane = threadIdx.x & 31;
    const int wave = threadIdx.x >> 5;
    const int m0   = blockIdx.x * 16;
    const int n0   = wave * 64;
    const int nl   = lane & 15;           // M index for A, N index for B/C
    const int hi   = lane >> 4;           // K-half selector (0 or 1)

    // ---- async-stage the contiguous 16 x Kdim A tile into LDS --------------
    {
        const float* gsrc = A + (size_t)m0 * Kdim;   // rows m0..m0+15, contiguous
        for (int idx = threadIdx.x * 4; idx < 16 * Kdim; idx += 128 * 4) {
            const int row = idx / Kdim;   // Kdim % 4 == 0: chunk never crosses rows
            const int col = idx - row * Kdim;
            async_copy16(gsrc + idx, &lds_a[row * lstr + col]);
        }
        wait_async0();
        __syncthreads();
    }

    // B layout (4x16 fp32): lane = N, K-half split across lane halves
    const float* b0 = W + (size_t)(n0 + nl) * Kdim + 2 * hi;
    const float* b1 = b0 + (size_t)16 * Kdim;
    const float* b2 = b0 + (size_t)32 * Kdim;
    const float* b3 = b0 + (size_t)48 * Kdim;

    // A fragment source in LDS: row = nl, element k + 2*hi
    const float* la = &lds_a[nl * lstr + 2 * hi];

    v8f acc[4];
    acc[0] = (v8f){0,0,0,0,0,0,0,0};
    acc[1] = acc[0]; acc[2] = acc[0]; acc[3] = acc[0];

    // prologue: stage B for K-step 0
    v2f w0 = *(const v2f*)(b0);
    v2f w1 = *(const v2f*)(b1);
    v2f w2 = *(const v2f*)(b2);
    v2f w3 = *(const v2f*)(b3);

#pragma unroll 2
    for (int k = 4; k < Kdim; k += 4) {
        // next step's B loads issued before consuming current registers
        v2f w0n = *(const v2f*)(b0 + k);
        v2f w1n = *(const v2f*)(b1 + k);
        v2f w2n = *(const v2f*)(b2 + k);
        v2f w3n = *(const v2f*)(b3 + k);

        v2f a = *(const v2f*)(la + (k - 4));   // low-latency LDS read
        acc[0] = WMMA_F32(a, w0, acc[0]);
        acc[1] = WMMA_F32(a, w1, acc[1]);
        acc[2] = WMMA_F32(a, w2, acc[2]);
        acc[3] = WMMA_F32(a, w3, acc[3]);

        w0 = w0n; w1 = w1n; w2 = w2n; w3 = w3n;
    }
    // epilogue
    {
        v2f a = *(const v2f*)(la + (Kdim - 4));
        acc[0] = WMMA_F32(a, w0, acc[0]);
        acc[1] = WMMA_F32(a, w1, acc[1]);
        acc[2] = WMMA_F32(a, w2, acc[2]);
        acc[3] = WMMA_F32(a, w3, acc[3]);
    }

    // C/D layout: VGPR r -> row m0 + r + 8*hi, col = n0 + 16t + nl
    const int rbase = hi * 8;
#pragma unroll
    for (int t = 0; t < 4; ++t) {
        const int col = n0 + 16 * t + nl;
        const float bv = bias ? bias[col] : 0.0f;
#pragma unroll
        for (int r = 0; r < 8; ++r) {
            C[(size_t)(m0 + rbase + r) * ldc + col] = acc[t][r] + bv;
        }
    }
}

// ---------------------------------------------------------------------------
// Edge-phase helpers
// ---------------------------------------------------------------------------
__device__ __forceinline__ void atomicMaxF(float* addr, float val)
{
    if (val >= 0.0f)
        atomicMax((int*)addr, __float_as_int(val));
    else
        atomicMin((unsigned int*)addr, __float_as_uint(val));
}

__device__ __forceinline__ void atomicAddF(float* addr, float val)
{
    __hip_atomic_fetch_add(addr, val, __ATOMIC_RELAXED, __HIP_MEMORY_SCOPE_AGENT);
}

__global__ void init_ms_kernel(float* __restrict__ m, float* __restrict__ s, int n)
{
    int i = blockIdx.x * blockDim.x + threadIdx.x;
    if (i < n) { m[i] = -__builtin_huge_valf(); s[i] = 0.0f; }
}

// alpha[e][h] = dot(q[dst], k[src]+e_proj) / sqrt(32); atomic segment max -> m
__global__ __launch_bounds__(256)
void alpha_max_kernel(const long long* __restrict__ ei,
                      const float* __restrict__ Q, const float* __restrict__ Kf,
                      const float* __restrict__ Ep,
                      float* __restrict__ alpha, float* __restrict__ m, int E)
{
    const int e = blockIdx.x * (blockDim.x >> 5) + (threadIdx.x >> 5);
    if (e >= E) return;
    const int lane = threadIdx.x & 31;
    const int src = (int)ei[e];
    const int dst = (int)ei[(size_t)E + e];

    const float* qrow = Q  + (size_t)dst * HC;
    const float* krow = Kf + (size_t)src * HC;
    const float* erow = Ep + (size_t)e   * HC;

    float p[HEADS];
#pragma unroll
    for (int j = 0; j < HEADS; ++j) {
        const int c = j * 32 + lane;
        p[j] = qrow[c] * (krow[c] + erow[c]);
    }
#pragma unroll
    for (int j = 0; j < HEADS; ++j) {
#pragma unroll
        for (int off = 16; off >= 1; off >>= 1)
            p[j] += __shfl_xor(p[j], off, 32);
    }
    // every lane holds all 8 head-sums; 3-level mux so lane j (<8) gets p[j]
    const float a0 = (lane & 1) ? p[1] : p[0];
    const float a1 = (lane & 1) ? p[3] : p[2];
    const float a2 = (lane & 1) ? p[5] : p[4];
    const float a3 = (lane & 1) ? p[7] : p[6];
    const float c0 = (lane & 2) ? a1 : a0;
    const float c1 = (lane & 2) ? a3 : a2;
    const float sel = ((lane & 4) ? c1 : c0) * ATT_SCALE;
    if (lane < HEADS) {
        alpha[(size_t)e * HEADS + lane] = sel;   // coalesced 8-wide store
        atomicMaxF(&m[dst * HEADS + lane], sel); // 8 parallel atomics
    }
}

// p = exp(alpha - m[dst]); alpha <- p; atomic segment sum -> s
__global__ void exp_sum_kernel(const long long* __restrict__ ei,
                               float* __restrict__ alpha,
                               const float* __restrict__ m,
                               float* __restrict__ s, int E)
{
    const long long t = (long long)blockIdx.x * blockDim.x + threadIdx.x;
    if (t >= (long long)E * HEADS) return;
    const int e = (int)(t >> 3), h = (int)(t & 7);
    const int dst = (int)ei[(size_t)E + e];
    const float pv = expf(alpha[t] - m[dst * HEADS + h]);
    alpha[t] = pv;
    atomicAddF(&s[dst * HEADS + h], pv);
}

// out[dst] += (v[src] + e_proj) * p/(s[dst]+1e-16)   (skip already in out)
__global__ __launch_bounds__(256)
void aggregate_kernel(const long long* __restrict__ ei,
                      const float* __restrict__ V, const float* __restrict__ Ep,
                      const float* __restrict__ palpha, const float* __restrict__ s,
                      float* __restrict__ out, int E)
{
    const int e = blockIdx.x * (blockDim.x >> 5) + (threadIdx.x >> 5);
    if (e >= E) return;
    const int lane = threadIdx.x & 31;
    const int src = (int)ei[e];
    const int dst = (int)ei[(size_t)E + e];

    const float* vrow = V  + (size_t)src * HC;
    const float* erow = Ep + (size_t)e   * HC;
    float*       orow = out + (size_t)dst * HC;

#pragma unroll
    for (int j = 0; j < HEADS; ++j) {
        const float w = palpha[(size_t)e * HEADS + j] / (s[dst * HEADS + j] + 1e-16f);
        const int c = j * 32 + lane;
        atomicAddF(&orow[c], (vrow[c] + erow[c]) * w);
    }
}

// ---------------------------------------------------------------------------
extern "C" void kernel_launch(void* const* d_in, const int* in_sizes, int n_in,
                              void* d_out, int out_size, void* d_ws, size_t ws_size,
                              hipStream_t stream)
{
    const float*     x   = (const float*)d_in[0];
    const long long* ei  = (const long long*)d_in[1];   // int64 edge_index [2,E]
    const float*     ea  = (const float*)d_in[2];
    const float*     Wq  = (const float*)d_in[3];
    const float*     bq  = (const float*)d_in[4];
    const float*     Wk  = (const float*)d_in[5];
    const float*     bk  = (const float*)d_in[6];
    const float*     Wv  = (const float*)d_in[7];
    const float*     bv  = (const float*)d_in[8];
    const float*     We  = (const float*)d_in[9];
    const float*     Wsk = (const float*)d_in[10];
    const float*     bsk = (const float*)d_in[11];
    float* out = (float*)d_out;

    const int N = in_sizes[0] / 256;        // 20000
    const int E = in_sizes[1] / 2;          // 320000

    // workspace partition (floats)
    float* ws    = (float*)d_ws;
    float* Q     = ws;
    float* Kf    = Q    + (size_t)N * HC;
    float* Vf    = Kf   + (size_t)N * HC;
    float* Ep    = Vf   + (size_t)N * HC;
    float* alpha = Ep   + (size_t)E * HC;
    float* mbuf  = alpha + (size_t)E * HEADS;
    float* sbuf  = mbuf + (size_t)N * HEADS;

    init_ms_kernel<<<(N * HEADS + 255) / 256, 256, 0, stream>>>(mbuf, sbuf, N * HEADS);

    gemm16_wmma<<<N / 16, 128, 0, stream>>>(x,  Wq,  bq,      Q,   256, HC);
    gemm16_wmma<<<N / 16, 128, 0, stream>>>(x,  Wk,  bk,      Kf,  256, HC);
    gemm16_wmma<<<N / 16, 128, 0, stream>>>(x,  Wv,  bv,      Vf,  256, HC);
    gemm16_wmma<<<N / 16, 128, 0, stream>>>(x,  Wsk, bsk,     out, 256, HC);  // skip -> out
    gemm16_wmma<<<E / 16, 128, 0, stream>>>(ea, We,  nullptr, Ep,  64,  HC);  // edge proj

    alpha_max_kernel<<<E / 8, 256, 0, stream>>>(ei, Q, Kf, Ep, alpha, mbuf, E);
    exp_sum_kernel<<<(E * HEADS) / 256, 256, 0, stream>>>(ei, alpha, mbuf, sbuf, E);
    aggregate_kernel<<<E / 8, 256, 0, stream>>>(ei, Vf, Ep, alpha, sbuf, out, E);
}